// DGLHGNN_27831388078174
// MI455X (gfx1250) — compile-verified
//
#include <hip/hip_runtime.h>
#include <math.h>

typedef float v2f __attribute__((ext_vector_type(2)));
typedef float v8f __attribute__((ext_vector_type(8)));

#define NV 50000
#define NE 20000
#define NNZ 500000
#define FIN 256

// ---------------------------------------------------------------------------
// Generic float zero-fill (grid-stride)
// ---------------------------------------------------------------------------
__global__ void hgnn_zero_f32(float* __restrict__ p, long n) {
    long i = (long)blockIdx.x * blockDim.x + threadIdx.x;
    long stride = (long)gridDim.x * blockDim.x;
    for (; i < n; i += stride) p[i] = 0.0f;
}

// ---------------------------------------------------------------------------
// Degree accumulation: deg_e[edge_idx[i]] += 1 ; deg_v[node_idx[i]] += 1
// ---------------------------------------------------------------------------
__global__ void hgnn_degrees(const int* __restrict__ node_idx,
                             const int* __restrict__ edge_idx,
                             float* __restrict__ degE,
                             float* __restrict__ degV, int nnz) {
    int i = blockIdx.x * blockDim.x + threadIdx.x;
    if (i < nnz) {
        atomicAdd(&degE[edge_idx[i]], 1.0f);
        atomicAdd(&degV[node_idx[i]], 1.0f);
    }
}

// inv[i] = 1 / max(deg[i], 1)
__global__ void hgnn_invdeg(const float* __restrict__ deg,
                            float* __restrict__ inv, int n) {
    int i = blockIdx.x * blockDim.x + threadIdx.x;
    if (i < n) inv[i] = 1.0f / fmaxf(deg[i], 1.0f);
}

// Zero-pad W3 [K x N] -> [K x Npad]
__global__ void hgnn_pad_w(const float* __restrict__ W, float* __restrict__ Wp,
                           int K, int N, int Npad) {
    int idx = blockIdx.x * blockDim.x + threadIdx.x;
    if (idx >= K * Npad) return;
    int k = idx / Npad, n = idx - k * Npad;
    Wp[idx] = (n < N) ? W[k * N + n] : 0.0f;
}

// ---------------------------------------------------------------------------
// FP32 WMMA GEMM: C[M,Ncols] = A[M,K] @ B[K,Npad] (row-major), using
// V_WMMA_F32_16X16X4_F32. One wave computes a 16x64 strip: the A fragment is
// loaded once per K-step and reused across 4 accumulators (4 wmma / iter).
//
// A-frag layout (16x4 f32): lanes 0-15 hold rows M=0..15; VGPR0 = K(0|2),
// VGPR1 = K(1|3) selected by lane>=16.  B-frag mirrors with N on lanes.
// C/D: VGPR r -> M = r + 8*(lane>=16), N = lane&15.
// ---------------------------------------------------------------------------
__global__ void hgnn_gemm_wmma_f32(const float* __restrict__ A,
                                   const float* __restrict__ B,
                                   float* __restrict__ C,
                                   int M, int K, int Npad, int Ncols) {
    const int wavesPerBlock = blockDim.x >> 5;
    const int wid     = blockIdx.x * wavesPerBlock + (threadIdx.x >> 5);
    const int ngroups = Npad >> 6;          // 64-wide column groups
    const int tm = wid / ngroups;
    const int tg = wid - tm * ngroups;
    if (tm * 16 >= M) return;

    const int lane  = threadIdx.x & 31;
    const int lhalf = lane >> 4;            // 0 | 1
    const int l16   = lane & 15;
    const int koff  = lhalf * 2;            // K sub-offset per A/B layout
    const int rowA  = tm * 16 + l16;
    const int colB  = tg * 64 + l16;

    const float* __restrict__ Arow = A + (size_t)rowA * K;

    v8f acc0 = {}, acc1 = {}, acc2 = {}, acc3 = {};

    for (int k0 = 0; k0 < K; k0 += 4) {
        v2f a = *(const v2f*)(Arow + k0 + koff);   // 8B-aligned: K even, k0%4==0

        const float* __restrict__ Bp0 = B + (size_t)(k0 + koff) * Npad + colB;
        const float* __restrict__ Bp1 = Bp0 + Npad;
        v2f b0 = { Bp0[0],  Bp1[0]  };
        v2f b1 = { Bp0[16], Bp1[16] };
        v2f b2 = { Bp0[32], Bp1[32] };
        v2f b3 = { Bp0[48], Bp1[48] };

        acc0 = __builtin_amdgcn_wmma_f32_16x16x4_f32(false, a, false, b0, (short)0, acc0, false, false);
        acc1 = __builtin_amdgcn_wmma_f32_16x16x4_f32(false, a, false, b1, (short)0, acc1, false, false);
        acc2 = __builtin_amdgcn_wmma_f32_16x16x4_f32(false, a, false, b2, (short)0, acc2, false, false);
        acc3 = __builtin_amdgcn_wmma_f32_16x16x4_f32(false, a, false, b3, (short)0, acc3, false, false);
    }

    const int rowBase = tm * 16 + lhalf * 8;
    const int c0 = tg * 64 + l16;
#pragma unroll
    for (int r = 0; r < 8; ++r) {
        float* Cp = C + (size_t)(rowBase + r) * Ncols + c0;
        if (c0 +  0 < Ncols) Cp[0]  = acc0[r];
        if (c0 + 16 < Ncols) Cp[16] = acc1[r];
        if (c0 + 32 < Ncols) Cp[32] = acc2[r];
        if (c0 + 48 < Ncols) Cp[48] = acc3[r];
    }
}

// ---------------------------------------------------------------------------
// Segment-sum scatter: dst[s[i], f] += src[g[i], f] * (SCALE ? scale[g[i]] : 1)
// grid.x = nnz, blockDim.x >= F (guarded). Traffic is L2-resident (192MB L2).
// ---------------------------------------------------------------------------
template <bool SCALE>
__global__ void hgnn_scatter(const float* __restrict__ src,
                             float* __restrict__ dst,
                             const int* __restrict__ g,
                             const int* __restrict__ s,
                             const float* __restrict__ scale, int F) {
    int f = threadIdx.x;
    if (f >= F) return;
    int i  = blockIdx.x;
    int gr = g[i];
    int sr = s[i];
    float v = src[(size_t)gr * F + f];
    if (SCALE) v *= scale[gr];
    atomicAdd(&dst[(size_t)sr * F + f], v);
}

// v[row,f] = relu( v[row,f] * invV[row] + b[f] )   grid.x = V, block = F
template <bool RELU>
__global__ void hgnn_finalize(float* __restrict__ v,
                              const float* __restrict__ invV,
                              const float* __restrict__ b, int F) {
    int row = blockIdx.x;
    int f   = threadIdx.x;
    if (f >= F) return;
    float x = v[(size_t)row * F + f] * invV[row] + b[f];
    if (RELU) x = fmaxf(x, 0.0f);
    v[(size_t)row * F + f] = x;
}

// ---------------------------------------------------------------------------
// Fused: x = vin*invV + b ; out = log_softmax(x) over F=40.
// One wave32 per row; lane handles elements {lane, lane+32}.
// ---------------------------------------------------------------------------
__global__ void hgnn_bias_logsoftmax(const float* __restrict__ vin,
                                     const float* __restrict__ invV,
                                     const float* __restrict__ b,
                                     float* __restrict__ out, int V, int F) {
    int row = blockIdx.x * (blockDim.x >> 5) + (threadIdx.x >> 5);
    if (row >= V) return;
    int lane = threadIdx.x & 31;
    float iv = invV[row];

    float x0 = -INFINITY, x1 = -INFINITY;
    if (lane < F)       x0 = vin[(size_t)row * F + lane]      * iv + b[lane];
    if (lane + 32 < F)  x1 = vin[(size_t)row * F + lane + 32] * iv + b[lane + 32];

    float m = fmaxf(x0, x1);
#pragma unroll
    for (int o = 16; o > 0; o >>= 1) m = fmaxf(m, __shfl_xor(m, o, 32));

    float sum = 0.0f;
    if (lane < F)      sum += expf(x0 - m);
    if (lane + 32 < F) sum += expf(x1 - m);
#pragma unroll
    for (int o = 16; o > 0; o >>= 1) sum += __shfl_xor(sum, o, 32);

    float ls = logf(sum);
    if (lane < F)      out[(size_t)row * F + lane]      = x0 - m - ls;
    if (lane + 32 < F) out[(size_t)row * F + lane + 32] = x1 - m - ls;
}

// ---------------------------------------------------------------------------
// Host orchestration
// ---------------------------------------------------------------------------
extern "C" void kernel_launch(void* const* d_in, const int* in_sizes, int n_in,
                              void* d_out, int out_size, void* d_ws, size_t ws_size,
                              hipStream_t stream) {
    const float* X  = (const float*)d_in[0];
    const int*   ni = (const int*)d_in[1];
    const int*   ei = (const int*)d_in[2];
    const float* W1 = (const float*)d_in[3];
    const float* b1 = (const float*)d_in[4];
    const float* W2 = (const float*)d_in[5];
    const float* b2 = (const float*)d_in[6];
    const float* W3 = (const float*)d_in[7];
    const float* b3 = (const float*)d_in[8];
    float* out = (float*)d_out;

    // Workspace carve-up (floats)
    float* ws   = (float*)d_ws;
    float* degE = ws;                            // 20000
    float* degV = degE + NE;                     // 50000
    float* invE = degV + NV;                     // 20000
    float* invV = invE + NE;                     // 50000
    float* H    = invV + NV;                     // NV*256
    float* EF   = H  + (size_t)NV * FIN;         // NE*256
    float* VF   = EF + (size_t)NE * FIN;         // NV*256
    float* H3   = VF + (size_t)NV * FIN;         // NV*40
    float* E3   = H3 + (size_t)NV * 40;          // NE*40
    float* V3   = E3 + (size_t)NE * 40;          // NV*40
    float* W3p  = V3 + (size_t)NV * 40;          // 256*64 zero-padded W3

    auto zero = [&](float* p, long n) {
        int blocks = (int)((n + 255) / 256);
        if (blocks > 65535) blocks = 65535;
        hipLaunchKernelGGL(hgnn_zero_f32, dim3(blocks), dim3(256), 0, stream, p, n);
    };

    // --- degrees ---
    zero(degE, NE + NV);
    hipLaunchKernelGGL(hgnn_degrees, dim3((NNZ + 255) / 256), dim3(256), 0, stream,
                       ni, ei, degE, degV, NNZ);
    hipLaunchKernelGGL(hgnn_invdeg, dim3((NE + NV + 255) / 256), dim3(256), 0, stream,
                       degE, invE, NE + NV);     // invE,invV contiguous like degE,degV

    // --- padded W3 (256x40 -> 256x64) ---
    hipLaunchKernelGGL(hgnn_pad_w, dim3((256 * 64 + 255) / 256), dim3(256), 0, stream,
                       W3, W3p, 256, 40, 64);

    auto gemm = [&](const float* A, const float* B, float* C, int Npad, int Ncols) {
        int waves  = (NV / 16) * (Npad / 64);
        int blocks = (waves + 7) / 8;            // 8 waves / 256-thread block
        hipLaunchKernelGGL(hgnn_gemm_wmma_f32, dim3(blocks), dim3(256), 0, stream,
                           A, B, C, NV, FIN, Npad, Ncols);
    };

    auto conv = [&](const float* in, const float* W, int Npad, int F,
                    float* Hb, float* Eb, float* Vb) {
        int fblk = (F + 63) & ~63;               // block = F rounded up to 64
        gemm(in, W, Hb, Npad, F);
        zero(Eb, (long)NE * F);
        hipLaunchKernelGGL((hgnn_scatter<false>), dim3(NNZ), dim3(fblk), 0, stream,
                           Hb, Eb, ni, ei, (const float*)nullptr, F);
        zero(Vb, (long)NV * F);
        hipLaunchKernelGGL((hgnn_scatter<true>), dim3(NNZ), dim3(fblk), 0, stream,
                           Eb, Vb, ei, ni, invE, F);
    };

    // Layer 1
    conv(X, W1, FIN, FIN, H, EF, VF);
    hipLaunchKernelGGL((hgnn_finalize<true>), dim3(NV), dim3(FIN), 0, stream,
                       VF, invV, b1, FIN);
    // Layer 2 (VF is consumed by GEMM before being re-zeroed/overwritten)
    conv(VF, W2, FIN, FIN, H, EF, VF);
    hipLaunchKernelGGL((hgnn_finalize<true>), dim3(NV), dim3(FIN), 0, stream,
                       VF, invV, b2, FIN);
    // Layer 3 (N=40, padded weights Npad=64)
    conv(VF, W3p, 64, 40, H3, E3, V3);
    hipLaunchKernelGGL(hgnn_bias_logsoftmax, dim3((NV + 7) / 8), dim3(256), 0, stream,
                       V3, invV, b3, out, NV, 40);
}